// CompetitiveLayer_58265526337832
// MI455X (gfx1250) — compile-verified
//
#include <hip/hip_runtime.h>

typedef __attribute__((ext_vector_type(2))) float v2f;
typedef __attribute__((ext_vector_type(8))) float v8f;

#define N      4096
#define BLOCK  256
#define WAVES  8                  // 256 threads / wave32
#define PAIRS  (N / 8 / WAVES)    // 64 chunk-pairs (8 k's each) per wave

// y[r] = tot[r] / (1 + sum_k M[r,k] * x[k]),  M = K (TRANS=0) or K^T (TRANS=1)
//
// One block handles 16 output rows; reduction runs through V_WMMA_F32_16X16X4_F32
// (full fp32). B = x-values replicated into all 16 columns, so every C column
// holds the 16 row partial sums; the accumulator chain does the k-reduction.
//
// Chunk-pair remap: one b128 A-load per lane covers K[row, k0+4h .. k0+4h+3].
// WMMA#1 takes the low pair (logical K = {k0,k0+1} lanes 0-15 / {k0+4,k0+5}
// lanes 16-31), WMMA#2 the high pair ({k0+2,k0+3}/{k0+6,k0+7}); the B fragments
// use the identical k-mapping, so the pair of WMMAs covers all 8 k's exactly once.
template <bool TRANS>
__global__ __launch_bounds__(BLOCK) void gemv_recip(const float* __restrict__ Km,
                                                    const float* __restrict__ x,
                                                    const float* __restrict__ tot,
                                                    float* __restrict__ y) {
    __shared__ float xs[N];            // 16 KB staged x vector
    __shared__ float part[WAVES * 16]; // per-wave partial row sums

    const int tid = threadIdx.x;

    // Cooperative stage of x into LDS (float4, fully coalesced)
    {
        const float4* x4 = (const float4*)x;
        float4* xs4 = (float4*)xs;
#pragma unroll
        for (int t = 0; t < N / 4 / BLOCK; ++t)
            xs4[tid + t * BLOCK] = x4[tid + t * BLOCK];
    }
    __syncthreads();

    const int wave = tid >> 5;
    const int lane = tid & 31;
    const int h    = lane >> 4;   // half-wave: logical K={0,1} vs K={2,3} slots
    const int l    = lane & 15;   // M index within tile
    const int r0   = blockIdx.x * 16;

    v8f c = {};                   // 16x16 f32 accumulator fragment
    const int kp0 = wave * PAIRS;

    if (!TRANS) {
        // A rows contiguous: one b128 per lane per chunk-pair
        const float* arow = Km + (size_t)(r0 + l) * N + 4 * h;
#pragma unroll 2
        for (int kp = 0; kp < PAIRS; ++kp) {
            const int k0 = 8 * (kp0 + kp);
            const float4 a4 = *(const float4*)(arow + k0);
            const float4 b4 = *(const float4*)(xs + k0 + 4 * h);
            v2f a1, a2, b1, b2;
            a1.x = a4.x; a1.y = a4.y;   // global k: {k0,k0+1} / {k0+4,k0+5}
            a2.x = a4.z; a2.y = a4.w;   // global k: {k0+2,k0+3} / {k0+6,k0+7}
            b1.x = b4.x; b1.y = b4.y;
            b2.x = b4.z; b2.y = b4.w;
            c = __builtin_amdgcn_wmma_f32_16x16x4_f32(false, a1, false, b1,
                                                      (short)0, c, false, false);
            c = __builtin_amdgcn_wmma_f32_16x16x4_f32(false, a2, false, b2,
                                                      (short)0, c, false, false);
        }
    } else {
        // A = K^T tile: column r0+l of K, rows along k (coalesced across lanes)
#pragma unroll 2
        for (int kp = 0; kp < PAIRS; ++kp) {
            const int k0 = 8 * (kp0 + kp);
            const float* p = Km + (size_t)(k0 + 4 * h) * N + (r0 + l);
            const float4 b4 = *(const float4*)(xs + k0 + 4 * h);
            v2f a1, a2, b1, b2;
            a1.x = p[0];
            a1.y = p[N];
            a2.x = p[2 * N];
            a2.y = p[3 * N];
            b1.x = b4.x; b1.y = b4.y;
            b2.x = b4.z; b2.y = b4.w;
            c = __builtin_amdgcn_wmma_f32_16x16x4_f32(false, a1, false, b1,
                                                      (short)0, c, false, false);
            c = __builtin_amdgcn_wmma_f32_16x16x4_f32(false, a2, false, b2,
                                                      (short)0, c, false, false);
        }
    }

    // C layout: VGPR r = row r (lanes 0-15) / row 8+r (lanes 16-31); columns equal.
    if (lane == 0) {
#pragma unroll
        for (int r = 0; r < 8; ++r) part[wave * 16 + r] = c[r];
    } else if (lane == 16) {
#pragma unroll
        for (int r = 0; r < 8; ++r) part[wave * 16 + 8 + r] = c[r];
    }
    __syncthreads();

    if (tid < 16) {
        float s = 0.0f;
#pragma unroll
        for (int w = 0; w < WAVES; ++w) s += part[w * 16 + tid];
        y[r0 + tid] = tot[r0 + tid] / (1.0f + s);
    }
}

// C[i,j] = K[i,j] * AF[i] * BF[j]   (float4 streaming)
__global__ __launch_bounds__(256) void outer_product_scale(const float* __restrict__ Km,
                                                           const float* __restrict__ AF,
                                                           const float* __restrict__ BF,
                                                           float* __restrict__ C) {
    const size_t g = (size_t)blockIdx.x * 256 + threadIdx.x; // one float4 per thread
    const float4 k4 = ((const float4*)Km)[g];
    const size_t e = g * 4;
    const int i = (int)(e >> 12);        // / 4096
    const int j = (int)(e & (size_t)(N - 1));
    const float af = AF[i];
    const float4 b4 = *(const float4*)(BF + j);
    float4 o;
    o.x = k4.x * af * b4.x;
    o.y = k4.y * af * b4.y;
    o.z = k4.z * af * b4.z;
    o.w = k4.w * af * b4.w;
    ((float4*)C)[g] = o;
}

extern "C" void kernel_launch(void* const* d_in, const int* in_sizes, int n_in,
                              void* d_out, int out_size, void* d_ws, size_t ws_size,
                              hipStream_t stream) {
    const float* AT = (const float*)d_in[0]; // [4096]
    const float* BT = (const float*)d_in[1]; // [4096]
    const float* K  = (const float*)d_in[2]; // [4096, 4096]
    float* out = (float*)d_out;              // [4096, 4096]

    float* AF = (float*)d_ws;                // [4096]
    float* BF = AF + N;                      // [4096]

    // BF starts as BT (graph-capture-safe async d2d copy)
    hipMemcpyAsync(BF, BT, N * sizeof(float), hipMemcpyDeviceToDevice, stream);

    for (int it = 0; it < 100; ++it) {
        gemv_recip<false><<<N / 16, BLOCK, 0, stream>>>(K, BF, AT, AF); // AF = AT/(1+K@BF)
        gemv_recip<true><<<N / 16, BLOCK, 0, stream>>>(K, AF, BT, BF);  // BF = BT/(1+K^T@AF)
    }

    outer_product_scale<<<(N / 4) * (N / 256), 256, 0, stream>>>(K, AF, BF, out);
}